// SGC_953482740186
// MI455X (gfx1250) — compile-verified
//
#include <hip/hip_runtime.h>
#include <hip/hip_bf16.h>

typedef __attribute__((ext_vector_type(2))) float v2f;
typedef __attribute__((ext_vector_type(8))) float v8f;

#define NNODES 100000
#define NFEAT  512
#define NCLASS 64
#define WT_STRIDE 516   // padded K-stride for transposed W in LDS (bank-conflict-free)

// ---------------------------------------------------------------------------
// 1) Edge normalization: w[e] = ew[e] * deg[row]^-0.5 * deg[col]^-0.5
// ---------------------------------------------------------------------------
__global__ void sgc_edge_norm(const long long* __restrict__ ei,
                              const float* __restrict__ ew,
                              const float* __restrict__ deg,
                              float* __restrict__ w, int E) {
    int e = blockIdx.x * blockDim.x + threadIdx.x;
    if (e >= E) return;
    int r = (int)ei[e];
    int c = (int)ei[(size_t)E + e];
    w[e] = ew[e] * __frsqrt_rn(deg[r]) * __frsqrt_rn(deg[c]);
}

// ---------------------------------------------------------------------------
// 2) GEMM h = x @ W via V_WMMA_F32_16X16X4_F32.
//    Block = 256 threads = 8 waves. W is staged TRANSPOSED in LDS
//    (ldsWT[c][k], stride 516 floats) so each B fragment {W[k][c],W[k+1][c]}
//    is a single contiguous ds_load_b64 into an even-aligned VGPR pair —
//    no register re-pairing movs between WMMAs, no LDS bank conflicts.
//    Each wave computes a 16x64 output tile.
// ---------------------------------------------------------------------------
__global__ void sgc_gemm_wmma(const float* __restrict__ x,
                              const float* __restrict__ wgt,
                              float* __restrict__ h, int nRowTiles) {
    extern __shared__ float ldsWT[];   // [NCLASS][WT_STRIDE] = 132KB

    // Cooperative transpose-load of W: global read coalesced, LDS write strided.
    for (int idx = threadIdx.x; idx < NFEAT * NCLASS; idx += 256) {
        int k = idx >> 6;          // 0..511
        int c = idx & 63;          // 0..63
        ldsWT[c * WT_STRIDE + k] = wgt[idx];
    }
    __syncthreads();

    const int wave = threadIdx.x >> 5;          // 0..7 (wave32)
    const int lane = threadIdx.x & 31;
    const int rowTile = blockIdx.x * 8 + wave;  // wave-uniform
    if (rowTile >= nRowTiles) return;           // whole wave exits together

    const int half = lane >> 4;                 // 0: lanes 0-15, 1: lanes 16-31
    const int l    = lane & 15;
    const int klo  = half * 2;                  // K offset held by this half
    const int row0 = rowTile * 16;

    const float* xrow = x + (size_t)(row0 + l) * NFEAT;   // A row for this lane
    const float* wcol = ldsWT + l * WT_STRIDE + klo;      // B col l, K-major

    v8f acc0 = {}, acc1 = {}, acc2 = {}, acc3 = {};

    #pragma unroll 4
    for (int k0 = 0; k0 < NFEAT; k0 += 4) {
        // A fragment: two consecutive K values -> one 8-byte global load
        v2f a = *(const v2f*)(xrow + k0 + klo);
        // B fragments: contiguous K pairs, 4 column tiles 16 apart
        v2f b0 = *(const v2f*)(wcol + k0);
        v2f b1 = *(const v2f*)(wcol + k0 + 16 * WT_STRIDE);
        v2f b2 = *(const v2f*)(wcol + k0 + 32 * WT_STRIDE);
        v2f b3 = *(const v2f*)(wcol + k0 + 48 * WT_STRIDE);

        acc0 = __builtin_amdgcn_wmma_f32_16x16x4_f32(false, a, false, b0,
                                                     (short)0, acc0, false, false);
        acc1 = __builtin_amdgcn_wmma_f32_16x16x4_f32(false, a, false, b1,
                                                     (short)0, acc1, false, false);
        acc2 = __builtin_amdgcn_wmma_f32_16x16x4_f32(false, a, false, b2,
                                                     (short)0, acc2, false, false);
        acc3 = __builtin_amdgcn_wmma_f32_16x16x4_f32(false, a, false, b3,
                                                     (short)0, acc3, false, false);
    }

    // Store D: VGPR i -> row = row0 + i + half*8, col = tile*16 + l
    #pragma unroll
    for (int i = 0; i < 8; ++i) {
        float* dst = h + (size_t)(row0 + i + half * 8) * NCLASS + l;
        dst[0]  = acc0[i];
        dst[16] = acc1[i];
        dst[32] = acc2[i];
        dst[48] = acc3[i];
    }
}

// ---------------------------------------------------------------------------
// 3) Zero fill
// ---------------------------------------------------------------------------
__global__ void sgc_fill_zero(float* __restrict__ p, int n) {
    int i = blockIdx.x * blockDim.x + threadIdx.x;
    if (i < n) p[i] = 0.0f;
}

// ---------------------------------------------------------------------------
// 4) Propagation: hout[col] += hin[row] * w  (edge-parallel scatter-add).
//    16 threads per edge, float4 gather (coalesced 256B row), scalar f32
//    atomics (global_atomic_add_f32, no return). h (25.6MB) is L2-resident.
// ---------------------------------------------------------------------------
__global__ void sgc_propagate(const long long* __restrict__ ei,
                              const float* __restrict__ w,
                              const float* __restrict__ hin,
                              float* __restrict__ hout, int E) {
    size_t tid = (size_t)blockIdx.x * blockDim.x + threadIdx.x;
    int e   = (int)(tid >> 4);
    if (e >= E) return;
    int sub = (int)(tid & 15);
    int r = (int)ei[e];
    int c = (int)ei[(size_t)E + e];
    float wv = w[e];
    const float4 v = *(const float4*)(hin + (size_t)r * NCLASS + sub * 4);
    float* dst = hout + (size_t)c * NCLASS + sub * 4;
    atomicAdd(dst + 0, v.x * wv);
    atomicAdd(dst + 1, v.y * wv);
    atomicAdd(dst + 2, v.z * wv);
    atomicAdd(dst + 3, v.w * wv);
}

// ---------------------------------------------------------------------------
// 5) Bias + log_softmax: one wave32 per row, 2 classes per lane,
//    __shfl_xor tree reductions (wave32-correct; no hardcoded 64-lane masks).
// ---------------------------------------------------------------------------
__global__ void sgc_log_softmax(const float* __restrict__ h,
                                const float* __restrict__ bias,
                                float* __restrict__ out, int n) {
    int row  = blockIdx.x * 8 + (threadIdx.x >> 5);
    if (row >= n) return;
    int lane = threadIdx.x & 31;
    const float* hr = h + (size_t)row * NCLASS;
    float z0 = hr[lane]      + bias[lane];
    float z1 = hr[32 + lane] + bias[32 + lane];

    float m = fmaxf(z0, z1);
    #pragma unroll
    for (int off = 16; off >= 1; off >>= 1)
        m = fmaxf(m, __shfl_xor(m, off, 32));

    float s = __expf(z0 - m) + __expf(z1 - m);
    #pragma unroll
    for (int off = 16; off >= 1; off >>= 1)
        s += __shfl_xor(s, off, 32);

    float lse = __logf(s) + m;
    float* orow = out + (size_t)row * NCLASS;
    orow[lane]      = z0 - lse;
    orow[32 + lane] = z1 - lse;
}

// ---------------------------------------------------------------------------
// Launch
// ---------------------------------------------------------------------------
extern "C" void kernel_launch(void* const* d_in, const int* in_sizes, int n_in,
                              void* d_out, int out_size, void* d_ws, size_t ws_size,
                              hipStream_t stream) {
    const float*     x    = (const float*)d_in[0];
    const float*     wgt  = (const float*)d_in[1];
    const float*     bias = (const float*)d_in[2];
    const long long* ei   = (const long long*)d_in[3];
    const float*     ew   = (const float*)d_in[4];
    const float*     deg  = (const float*)d_in[5];
    float*           out  = (float*)d_out;

    const int E = in_sizes[4];          // edge_weight count = E
    const int n = NNODES;
    const int hElems = n * NCLASS;

    char* ws = (char*)d_ws;
    size_t off = 0;
    float* wNorm = (float*)(ws + off);  off += ((size_t)E * 4 + 255) & ~(size_t)255;
    float* hA    = (float*)(ws + off);  off += ((size_t)hElems * 4 + 255) & ~(size_t)255;
    float* hB    = (float*)(ws + off);

    // 1) normalized edge weights
    sgc_edge_norm<<<(E + 255) / 256, 256, 0, stream>>>(ei, ew, deg, wNorm, E);

    // 2) hA = x @ W  (WMMA, W transposed into 132KB dynamic LDS)
    const int nRowTiles = n / 16;                       // 6250
    const int gemmBlocks = (nRowTiles + 7) / 8;         // 782
    sgc_gemm_wmma<<<gemmBlocks, 256, NCLASS * WT_STRIDE * sizeof(float), stream>>>(
        x, wgt, hA, nRowTiles);

    // 3) round 1: hB = scatter(hA)
    sgc_fill_zero<<<(hElems + 255) / 256, 256, 0, stream>>>(hB, hElems);
    {
        size_t work = (size_t)E * 16;
        sgc_propagate<<<(unsigned)((work + 255) / 256), 256, 0, stream>>>(ei, wNorm, hA, hB, E);
    }

    // 4) round 2: hA = scatter(hB)
    sgc_fill_zero<<<(hElems + 255) / 256, 256, 0, stream>>>(hA, hElems);
    {
        size_t work = (size_t)E * 16;
        sgc_propagate<<<(unsigned)((work + 255) / 256), 256, 0, stream>>>(ei, wNorm, hB, hA, E);
    }

    // 5) out = log_softmax(hA + bias)
    sgc_log_softmax<<<(n + 7) / 8, 256, 0, stream>>>(hA, bias, out, n);
}